// OrthogonalDense_10024453669625
// MI455X (gfx1250) — compile-verified
//
#include <hip/hip_runtime.h>

// y = x @ (H0 H1 ... H7) @ diag(d) + bias, H_i = I - 2 v_i v_i^T (v_i normalized)
// Collapsed to 8 sequential rank-1 reflections applied to rows of x:
//   z <- z - 2 (z . vn_k) vn_k,  k = 0..7;  y = z * d + bias
// Memory-bound: read x once (64MB, non-temporal), write y once (64MB,
// non-temporal). One row per wave32, row kept entirely in VGPRs (16 x f32x4
// per lane). vn (64KB) broadcast to LDS once per workgroup via the Tensor
// Data Mover (tensor_load_to_lds), overlapped with the row fetch.

#define NFEAT 2048
#define ORDER 8
#define BATCH 8192
#define ROWS_PER_BLOCK 8
#define CHUNKS 16 // 2048 cols / (32 lanes * 4 floats)

typedef __attribute__((ext_vector_type(4))) float f32x4;
typedef __attribute__((ext_vector_type(4))) unsigned int u32x4_t;
typedef __attribute__((ext_vector_type(8))) int i32x8_t;
typedef __attribute__((ext_vector_type(4))) int i32x4_t;

// TDM builtins exist only in the device compilation pass; the host pass must
// still parse the kernel body, so give it the plain-copy fallback. On the
// DEVICE pass, missing builtins are a hard error (no silent fallback).
#if defined(__has_builtin) && __has_builtin(__builtin_amdgcn_tensor_load_to_lds) && \
    __has_builtin(__builtin_amdgcn_s_wait_tensorcnt)
#define USE_TDM 1
#else
#define USE_TDM 0
#if defined(__HIP_DEVICE_COMPILE__)
#error "gfx1250 TDM builtins unavailable in device compilation"
#endif
#endif

__device__ __forceinline__ float wave_reduce_add(float v) {
#pragma unroll
  for (int off = 16; off > 0; off >>= 1) v += __shfl_xor(v, off, 32);
  return v;
}

// ---------------------------------------------------------------------------
// Main kernel: apply 8 Householder reflections + diag(d) + bias.
// 256 threads = 8 waves; one row (2048 f32) per wave, held in 16 f32x4 VGPRs.
// vn (8 x 2048 f32 = 64KB) DMA'd into LDS once per block via TDM.
// ---------------------------------------------------------------------------
__global__ void __launch_bounds__(256) hh_apply(const float* __restrict__ x,
                                                const float* __restrict__ vn,
                                                const float* __restrict__ dvec,
                                                const float* __restrict__ bias,
                                                float* __restrict__ y) {
  extern __shared__ float vs[]; // ORDER * NFEAT floats (64 KB)

#if USE_TDM
  // --- Issue TDM first so the 64KB LDS broadcast overlaps the x-row loads ---
  // One TDM descriptor: 1-D tile of 16384 f32 elements, global vn -> LDS.
  // D# layout per CDNA5 ISA (group0 128b, group1 256b).
  if (threadIdx.x < 32) { // wave 0 issues; TDM ignores EXEC, tracked by TENSORcnt
    const unsigned lds_off = (unsigned)(size_t)vs;          // low 32 bits = LDS byte offset
    const unsigned long long ga = (unsigned long long)(size_t)vn;
    u32x4_t g0 = {
        1u,                                                 // count=1 (valid), user mode
        lds_off,                                            // lds_addr [63:32]
        (unsigned)ga,                                       // global_addr[31:0]
        (((unsigned)(ga >> 32)) & 0x01FFFFFFu) | (2u << 30) // global_addr[56:32] | type=2
    };
    i32x8_t g1 = {
        0x00020000,          // workgroup_mask=0, data_size=2 (4 bytes)
        (int)(16384u << 16), // tensor_dim0[15:0]=16384 in bits[63:48]
        0x00010000,          // tensor_dim0[31:16]=0, tensor_dim1=1
        (int)(16384u << 16), // tensor_dim1 hi=0, tile_dim0=16384
        0,                   // tile_dim1=0, tile_dim2=0 (unused -> 1-D tile)
        16384,               // tensor_dim0_stride low 32
        0, 0                 // stride hi / dim1 stride = 0
    };
    i32x4_t g2 = {0, 0, 0, 0};
    i32x4_t g3 = {0, 0, 0, 0};
#if __clang_major__ >= 23
    i32x8_t g4 = {0, 0, 0, 0, 0, 0, 0, 0};
    __builtin_amdgcn_tensor_load_to_lds(g0, g1, g2, g3, g4, 0);
#else
    __builtin_amdgcn_tensor_load_to_lds(g0, g1, g2, g3, 0);
#endif
  }
#endif

  const int wave = threadIdx.x >> 5;
  const int lane = threadIdx.x & 31;
  const size_t row = (size_t)blockIdx.x * ROWS_PER_BLOCK + wave;

  // Load row while TDM runs. Chunk c covers cols [c*128, c*128+128); lane holds
  // cols c*128 + 4*lane ..+3. Non-temporal: x is touched exactly once.
  const f32x4* xr = (const f32x4*)(x + row * NFEAT);
  f32x4 z[CHUNKS];
#pragma unroll
  for (int c = 0; c < CHUNKS; ++c) z[c] = __builtin_nontemporal_load(xr + c * 32 + lane);

#if USE_TDM
  __builtin_amdgcn_s_wait_tensorcnt(0); // wave 0 waits for TDM; trivial for others
#else
  { // host-pass placeholder (and safety net): cooperative copy vn -> LDS
    const f32x4* src = (const f32x4*)vn;
    f32x4* dst = (f32x4*)vs;
    for (int i = threadIdx.x; i < (ORDER * NFEAT) / 4; i += 256) dst[i] = src[i];
  }
#endif
  __syncthreads();

  for (int k = 0; k < ORDER; ++k) {
    const f32x4* vk = (const f32x4*)(vs + k * NFEAT);
    f32x4 vv[CHUNKS];
    float ax = 0.f, ay = 0.f, az = 0.f, aw = 0.f;
#pragma unroll
    for (int c = 0; c < CHUNKS; ++c) {
      vv[c] = vk[c * 32 + lane]; // ds_load_b128, conflict-free
      ax = fmaf(z[c].x, vv[c].x, ax);
      ay = fmaf(z[c].y, vv[c].y, ay);
      az = fmaf(z[c].z, vv[c].z, az);
      aw = fmaf(z[c].w, vv[c].w, aw);
    }
    const float acc = wave_reduce_add((ax + ay) + (az + aw)); // full-row dot
    const float t = -2.f * acc;
#pragma unroll
    for (int c = 0; c < CHUNKS; ++c) { // z -= 2 (z.v) v, v reused from VGPRs
      z[c].x = fmaf(t, vv[c].x, z[c].x);
      z[c].y = fmaf(t, vv[c].y, z[c].y);
      z[c].z = fmaf(t, vv[c].z, z[c].z);
      z[c].w = fmaf(t, vv[c].w, z[c].w);
    }
  }

  // Epilogue: y = z * d + bias (d/bias cached: reused by every block)
  const f32x4* dv = (const f32x4*)dvec;
  const f32x4* bv = (const f32x4*)bias;
  f32x4* yr = (f32x4*)(y + row * NFEAT);
#pragma unroll
  for (int c = 0; c < CHUNKS; ++c) {
    const f32x4 dd = dv[c * 32 + lane];
    const f32x4 bb = bv[c * 32 + lane];
    f32x4 o;
    o.x = fmaf(z[c].x, dd.x, bb.x);
    o.y = fmaf(z[c].y, dd.y, bb.y);
    o.z = fmaf(z[c].z, dd.z, bb.z);
    o.w = fmaf(z[c].w, dd.w, bb.w);
    __builtin_nontemporal_store(o, yr + c * 32 + lane); // y written exactly once
  }
}

// ---------------------------------------------------------------------------
// Kernel 1: vn[k] = v[k] / ||v[k]||  (8 blocks, one per reflection vector)
// ---------------------------------------------------------------------------
__global__ void __launch_bounds__(256) hh_normalize_v(const float* __restrict__ v,
                                                      float* __restrict__ vn) {
  __shared__ float red[8];
  __shared__ float s_inv;
  const int k = blockIdx.x;
  const float* row = v + (size_t)k * NFEAT;

  float t[NFEAT / 256]; // 8 elements per thread, fixed trip counts
  float s = 0.f;
#pragma unroll
  for (int j = 0; j < NFEAT / 256; ++j) {
    t[j] = row[threadIdx.x + j * 256];
    s = fmaf(t[j], t[j], s);
  }
  s = wave_reduce_add(s);
  const int wave = threadIdx.x >> 5;
  const int lane = threadIdx.x & 31;
  if (lane == 0) red[wave] = s;
  __syncthreads();
  if (wave == 0) {
    float p = (lane < 8) ? red[lane] : 0.f;
    p = wave_reduce_add(p);
    if (lane == 0) s_inv = 1.f / sqrtf(p);
  }
  __syncthreads();
  const float inv = s_inv;
#pragma unroll
  for (int j = 0; j < NFEAT / 256; ++j)
    vn[(size_t)k * NFEAT + threadIdx.x + j * 256] = t[j] * inv;
}

extern "C" void kernel_launch(void* const* d_in, const int* in_sizes, int n_in,
                              void* d_out, int out_size, void* d_ws, size_t ws_size,
                              hipStream_t stream) {
  const float* x = (const float*)d_in[0];
  const float* v = (const float*)d_in[1];
  const float* dvec = (const float*)d_in[2];
  const float* bias = (const float*)d_in[3];
  float* y = (float*)d_out;
  float* vn = (float*)d_ws; // ORDER*NFEAT floats = 64 KB scratch

  hh_normalize_v<<<ORDER, 256, 0, stream>>>(v, vn);
  hh_apply<<<BATCH / ROWS_PER_BLOCK, 256, ORDER * NFEAT * sizeof(float), stream>>>(
      x, vn, dvec, bias, y);
}